// GlobalMemoryModule_75539884802825
// MI455X (gfx1250) — compile-verified
//
#include <hip/hip_runtime.h>

typedef float v2f __attribute__((ext_vector_type(2)));
typedef float v8f __attribute__((ext_vector_type(8)));

#define S_ 5
#define B_ 8
#define H_ 512
#define W_ 512
#define C_ 64
#define HP_ 257
#define WP_ 257
#define NPIX (HP_*WP_)              // 66049 pixels per (s,b) image
#define NTILES ((NPIX + 15) / 16)   // 4129 pixel tiles (last one ragged)
#define TPW 16                      // pixel tiles per wave (amortizes setup/A)
#define NCHUNK ((NTILES + TPW - 1) / TPW)   // 259 chunks
#define EPS_ 1e-5f

// ---------------------------------------------------------------------------
// Kernel 1: separable 2-tap DWT (pad 1, stride 2) -> L and Hc, [S,B,1,257,257]
// Memory-bound stencil; x (8 MB) stays L2-resident across the 5 scales.
// ---------------------------------------------------------------------------
__global__ __launch_bounds__(256) void dwt_kernel(
    const float* __restrict__ x, const float* __restrict__ lp,
    const float* __restrict__ hp,
    float* __restrict__ Lout, float* __restrict__ Hout)
{
    int tid = blockIdx.x * blockDim.x + threadIdx.x;
    if (tid >= S_ * B_ * NPIX) return;

    int hw = tid % NPIX;
    int t2 = tid / NPIX;
    int b  = t2 & 7;
    int s  = t2 >> 3;
    int hh = hw / WP_;
    int ww = hw % WP_;

    // L[h',w'] = sum_{i,j in {0,1}} k[i]*k[j] * x[2h'-1+i][2w'-1+j] (0 if OOB)
    int r0 = 2 * hh - 1, r1 = 2 * hh;
    int c0 = 2 * ww - 1, c1 = 2 * ww;
    const float* xb = x + (size_t)b * H_ * W_;

    float x00 = 0.f, x01 = 0.f, x10 = 0.f, x11 = 0.f;
    bool r0v = (r0 >= 0), r1v = (r1 < H_);
    bool c0v = (c0 >= 0), c1v = (c1 < W_);
    if (r0v && c0v) x00 = xb[r0 * W_ + c0];
    if (r0v && c1v) x01 = xb[r0 * W_ + c1];
    if (r1v && c0v) x10 = xb[r1 * W_ + c0];
    if (r1v && c1v) x11 = xb[r1 * W_ + c1];

    float l0 = lp[2 * s], l1 = lp[2 * s + 1];
    float h0 = hp[2 * s], h1 = hp[2 * s + 1];

    float Lv = l0 * l0 * x00 + l0 * l1 * x01 + l1 * l0 * x10 + l1 * l1 * x11;
    float Hv = h0 * h0 * x00 + h0 * h1 * x01 + h1 * h0 * x10 + h1 * h1 * x11;

    int oidx = (s * B_ + b) * NPIX + hw;
    Lout[oidx] = Lv;
    Hout[oidx] = Hv;
}

// ---------------------------------------------------------------------------
// Kernel 2: fused (1->64 lift + BN1 + ReLU) -> (64x64 mix via WMMA) -> BN2+ReLU
// One wave per (scale, batch, o-tile, chunk of TPW pixel tiles).
// A = s2[o]*W2[o,c] resident in 32 VGPRs, (a1,sh1) pairs resident in 64 VGPRs;
// steady-state loop: 1 unconditional (clamped) L load + 16 x (FMA+MAX+WMMA)
// + 8 masked stores per tile. EXEC stays all-1s through the WMMA chain.
// ---------------------------------------------------------------------------
__global__ __launch_bounds__(128) void fused_gemm_kernel(
    const float* __restrict__ w1, const float* __restrict__ w2,
    const float* __restrict__ g1, const float* __restrict__ b1,
    const float* __restrict__ m1, const float* __restrict__ v1,
    const float* __restrict__ g2, const float* __restrict__ b2,
    const float* __restrict__ m2, const float* __restrict__ v2,
    const float* __restrict__ Lbuf, float* __restrict__ Pout)
{
    // Per-wave LDS slots (4 waves / block), broadcast-read constants.
    __shared__ __align__(16) float4 pairsh[4][32];  // (a1[2j],a1[2j+1],sh1[2j],sh1[2j+1])
    __shared__ __align__(16) float  s2sh[4][64];    // g2/sqrt(v2+eps)
    __shared__ __align__(16) float  sh2sh[4][64];   // b2 - m2*s2

    int lane  = threadIdx.x & 31;
    int wslot = threadIdx.x >> 5;
    int wid   = blockIdx.x * 4 + wslot;

    int otile = wid & 3;
    int r     = wid >> 2;
    int chunk = r % NCHUNK;
    int r2    = r / NCHUNK;
    int b     = r2 & 7;
    int s     = r2 >> 3;
    if (s >= S_) return;   // grid is exact; defensive only

    // ---- fold BN constants (each lane covers 2 channels) ----
    {
        int base = s * C_;
        int ca = 2 * lane, cb = 2 * lane + 1;
        float sca = g1[base + ca] / sqrtf(v1[base + ca] + EPS_);
        float scb = g1[base + cb] / sqrtf(v1[base + cb] + EPS_);
        float4 pr;
        pr.x = w1[base + ca] * sca;                   // a1 even
        pr.y = w1[base + cb] * scb;                   // a1 odd
        pr.z = b1[base + ca] - m1[base + ca] * sca;   // sh1 even
        pr.w = b1[base + cb] - m1[base + cb] * scb;   // sh1 odd
        pairsh[wslot][lane] = pr;

        int oa = lane, ob = lane + 32;
        float s2a = g2[base + oa] / sqrtf(v2[base + oa] + EPS_);
        float s2b = g2[base + ob] / sqrtf(v2[base + ob] + EPS_);
        s2sh[wslot][oa]  = s2a;
        s2sh[wslot][ob]  = s2b;
        sh2sh[wslot][oa] = b2[base + oa] - m2[base + oa] * s2a;
        sh2sh[wslot][ob] = b2[base + ob] - m2[base + ob] * s2b;
    }
    __syncthreads();

    // ---- load resident A tiles: A[m][k] = s2[o]*W2[o][c] ----
    // 16x4 f32 A layout: lane<16 -> (M=lane, K=0,1); lane>=16 -> (M=lane-16, K=2,3)
    int orow  = otile * 16 + (lane & 15);
    int chalf = (lane >> 4) << 1;            // 0 or 2
    float s2v = s2sh[wslot][orow];
    const float* w2row = w2 + ((size_t)(s * C_ + orow)) * C_;
    v2f A[16];
#pragma unroll
    for (int kk = 0; kk < 16; ++kk) {
        float2 wv = *(const float2*)(w2row + kk * 4 + chalf);
        A[kk].x = wv.x * s2v;
        A[kk].y = wv.y * s2v;
    }

    // ---- hoist (a1,sh1) channel-pair constants into registers ----
    // B 4x16 layout needs channels c = 4kk + 2*(lane>=16) + {0,1}; pair j = 2kk + (lane>>4)
    float4 prs[16];
#pragma unroll
    for (int kk = 0; kk < 16; ++kk)
        prs[kk] = pairsh[wslot][(kk << 1) | (lane >> 4)];

    // shift2 for epilogue: C/D layout rows are v (+8 for upper half-wave)
    int obase = otile * 16 + ((lane >> 4) << 3);
    float4 shlo = *(const float4*)&sh2sh[wslot][obase];
    float4 shhi = *(const float4*)&sh2sh[wslot][obase + 4];
    float sh[8] = {shlo.x, shlo.y, shlo.z, shlo.w,
                   shhi.x, shhi.y, shhi.z, shhi.w};

    // ---- pixel-tile loop over this wave's chunk ----
    int pixcol = lane & 15;
    int tile0  = chunk * TPW;
    int ntile  = NTILES - tile0;
    if (ntile > TPW) ntile = TPW;
    const float* Lrow  = Lbuf + (size_t)(s * B_ + b) * NPIX;
    float*       Prow0 = Pout + ((size_t)(s * B_ + b) * C_ + obase) * NPIX;

    for (int t = 0; t < ntile; ++t) {
        int pix    = (tile0 + t) * 16 + pixcol;
        bool valid = pix < NPIX;                  // only last tile is ragged
        // Clamped, unconditional load: no EXEC manipulation in the hot loop;
        // invalid lanes compute a duplicate pixel and are masked at the store.
        int pixc   = valid ? pix : (NPIX - 1);
        float Lv   = Lrow[pixc];

        v8f acc = {};
#pragma unroll
        for (int kk = 0; kk < 16; ++kk) {
            float4 pr = prs[kk];
            v2f bvec;
            bvec.x = fmaxf(fmaf(pr.x, Lv, pr.z), 0.0f);
            bvec.y = fmaxf(fmaf(pr.y, Lv, pr.w), 0.0f);
            acc = __builtin_amdgcn_wmma_f32_16x16x4_f32(
                /*neg_a=*/false, A[kk], /*neg_b=*/false, bvec,
                /*c_mod=*/(short)0, acc, /*reuse_a=*/false, /*reuse_b=*/false);
        }

        // BN2 bias + ReLU, strided coalesced stores of P (masked on ragged tile)
        if (valid) {
            float* pbase = Prow0 + pix;
#pragma unroll
            for (int v = 0; v < 8; ++v) {
                pbase[(size_t)v * NPIX] = fmaxf(acc[v] + sh[v], 0.0f);
            }
        }
    }
}

// ---------------------------------------------------------------------------
extern "C" void kernel_launch(void* const* d_in, const int* in_sizes, int n_in,
                              void* d_out, int out_size, void* d_ws, size_t ws_size,
                              hipStream_t stream) {
    (void)in_sizes; (void)n_in; (void)out_size; (void)d_ws; (void)ws_size;

    const float* x  = (const float*)d_in[0];
    const float* lp = (const float*)d_in[1];
    const float* hp = (const float*)d_in[2];
    const float* w1 = (const float*)d_in[3];
    const float* w2 = (const float*)d_in[4];
    const float* g1 = (const float*)d_in[5];
    const float* b1 = (const float*)d_in[6];
    const float* m1 = (const float*)d_in[7];
    const float* v1 = (const float*)d_in[8];
    const float* g2 = (const float*)d_in[9];
    const float* b2 = (const float*)d_in[10];
    const float* m2 = (const float*)d_in[11];
    const float* v2 = (const float*)d_in[12];

    float* out = (float*)d_out;
    const size_t SL = (size_t)S_ * B_ * NPIX;   // elements of L (and of Hc)
    float* Lout = out;
    float* Hout = out + SL;
    float* Pout = out + 2 * SL;

    {
        int total  = S_ * B_ * NPIX;
        int blocks = (total + 255) / 256;
        dwt_kernel<<<blocks, 256, 0, stream>>>(x, lp, hp, Lout, Hout);
    }
    {
        int waves  = S_ * B_ * NCHUNK * 4;   // 41,440 waves
        int blocks = waves / 4;              // exact: 10,360 blocks of 4 waves
        fused_gemm_kernel<<<blocks, 128, 0, stream>>>(
            w1, w2, g1, b1, m1, v1, g2, b2, m2, v2, Lout, Pout);
    }
}